// EVGNetwork_18159121728072
// MI455X (gfx1250) — compile-verified
//
#include <hip/hip_runtime.h>
#include <math.h>
#include <float.h>

// Problem dims (match reference)
#define D_IN  512
#define D_HID 128
#define D_OUT 512
#define N_ENT 262144

#define NB_LOGITS 1024  // blocks for streaming logits kernel: 8192 waves x 2 tiles = 16384 tiles
#define NB_SUM    512   // blocks for exp-sum + candidate kernel (2 logits per thread)
#define NCAND     8     // per-block top-candidates kept (>= max supported k)

typedef float v2f __attribute__((ext_vector_type(2)));
typedef float v8f __attribute__((ext_vector_type(8)));

// Workspace layout (float offsets). Total ~272K floats ~= 1.09 MB.
#define O_LOGITS 0
#define O_KQ     (N_ENT)                        // 512: (Wk^T q)/sqrt(H)
#define O_C0     (N_ENT + D_IN)                 // 1:   (bk.q)/sqrt(H)
#define O_BMAX   (N_ENT + D_IN + 32)            // NB_LOGITS
#define O_BSUM   (O_BMAX + NB_LOGITS)           // NB_SUM
#define O_GMAX   (O_BSUM + NB_SUM)              // 1
#define O_CANDV  (O_GMAX + 32)                  // NB_SUM*NCAND floats
#define O_CANDI  (O_CANDV + NB_SUM * NCAND)     // NB_SUM*NCAND ints (reinterpreted)

// ---------------------------------------------------------------------------
// Kernel 1: tiny prep. q = Wq c + bq ; kq = (Wk^T q)/sqrt(H) ; c0 = (bk.q)/sqrt(H)
// ---------------------------------------------------------------------------
__global__ void evg_prep(const float* __restrict__ cemb,
                         const float* __restrict__ Wq, const float* __restrict__ bq,
                         const float* __restrict__ Wk, const float* __restrict__ bk,
                         float* __restrict__ ws)
{
    __shared__ float sq[D_HID];
    const int tid = threadIdx.x;                  // 0..511
    if (tid < D_HID) {
        float acc = bq[tid];
        const float* wr = Wq + tid * D_IN;
        for (int d = 0; d < D_IN; ++d) acc += wr[d] * cemb[d];
        sq[tid] = acc;
    }
    __syncthreads();
    const float invs = 0.088388347648318447f;     // 1/sqrt(128)
    float acc = 0.f;
    for (int h = 0; h < D_HID; ++h) acc += Wk[h * D_IN + tid] * sq[h];
    ws[O_KQ + tid] = invs * acc;
    if (tid == 0) {
        float c0 = 0.f;
        for (int h = 0; h < D_HID; ++h) c0 += bk[h] * sq[h];
        ws[O_C0] = invs * c0;
    }
}

// ---------------------------------------------------------------------------
// Kernel 2: dominant streaming kernel (512 MB @ 23.3 TB/s => ~22 us floor).
// logits[i] = E[i,:].kq + c0 via V_WMMA_F32_16X16X4_F32.
// Each wave owns TWO 16-entity tiles: one shared A (kq chunk from LDS) feeds
// two INDEPENDENT WMMA accumulator chains -> no back-to-back RAW on the XDL
// pipe, 2 clause-grouped b64 loads per k-quad. 8192 waves, exactly 2 tiles each.
// ---------------------------------------------------------------------------
__global__ void evg_logits(const float* __restrict__ E, float* __restrict__ ws)
{
    __shared__ __align__(16) float s_kq[D_IN];
    __shared__ float sred[256];
    const int tid = threadIdx.x;                  // 0..255
    s_kq[tid]       = ws[O_KQ + tid];
    s_kq[tid + 256] = ws[O_KQ + tid + 256];
    __syncthreads();
    const float c0 = ws[O_C0];

    const float2* skq2 = reinterpret_cast<const float2*>(s_kq);
    const int lane = tid & 31;
    const int half = lane >> 4;                   // 0: K pair {0,1}, 1: K pair {2,3}
    const int r16  = lane & 15;                   // entity within tile
    const int waveId = (blockIdx.x * blockDim.x + tid) >> 5;   // 0..8191
    const int t0 = waveId * 2;                    // tiles (N_ENT/16 = 16384 total)
    const int t1 = t0 + 1;

    const float2* rp0 = reinterpret_cast<const float2*>(E + (size_t)(t0 * 16 + r16) * D_IN);
    const float2* rp1 = reinterpret_cast<const float2*>(E + (size_t)(t1 * 16 + r16) * D_IN);

    v8f ca = {0.f,0.f,0.f,0.f,0.f,0.f,0.f,0.f};
    v8f cb = {0.f,0.f,0.f,0.f,0.f,0.f,0.f,0.f};
#pragma unroll 4
    for (int k0 = 0; k0 < D_IN; k0 += 4) {
        const float2 aa = skq2[(k0 >> 1) + half]; // LDS broadcast, shared by both tiles
        const float2 b0 = rp0 [(k0 >> 1) + half]; // streaming global_load_b64
        const float2 b1 = rp1 [(k0 >> 1) + half];
        v2f a  = {aa.x, aa.y};
        v2f v0 = {b0.x, b0.y};
        v2f v1 = {b1.x, b1.y};
        ca = __builtin_amdgcn_wmma_f32_16x16x4_f32(false, a, false, v0, (short)0, ca, false, false);
        cb = __builtin_amdgcn_wmma_f32_16x16x4_f32(false, a, false, v1, (short)0, cb, false, false);
    }
    const float l0 = ca[0] + c0;                  // every D row identical; c[0] = logit
    const float l1 = cb[0] + c0;
    float lmax = fmaxf(l0, l1);
    if (lane < 16) {
        ws[O_LOGITS + t0 * 16 + lane] = l0;
        ws[O_LOGITS + t1 * 16 + lane] = l1;
    }

    sred[tid] = lmax;
    __syncthreads();
    for (int s = 128; s > 0; s >>= 1) {
        if (tid < s) sred[tid] = fmaxf(sred[tid], sred[tid + s]);
        __syncthreads();
    }
    if (tid == 0) ws[O_BMAX + blockIdx.x] = sred[0];
}

// ---------------------------------------------------------------------------
// Kernel 3: reduce 1024 block maxes -> global max. 1 block x 512 threads.
// ---------------------------------------------------------------------------
__global__ void evg_gmax(float* __restrict__ ws)
{
    __shared__ float sred[512];
    const int tid = threadIdx.x;
    sred[tid] = fmaxf(ws[O_BMAX + tid], ws[O_BMAX + tid + 512]);
    __syncthreads();
    for (int s = 256; s > 0; s >>= 1) {
        if (tid < s) sred[tid] = fmaxf(sred[tid], sred[tid + s]);
        __syncthreads();
    }
    if (tid == 0) ws[O_GMAX] = sred[0];
}

// ---------------------------------------------------------------------------
// Kernel 4: per-block exp-sum partials AND per-block top-NCAND candidates.
// 512 blocks x 256 threads: each thread owns exactly 2 logits (in registers).
// Candidate merge is exact for k<=NCAND under (value, -index) lexicographic
// order. Deterministic: tree reductions only, no float atomics.
// ---------------------------------------------------------------------------
__global__ void evg_expsum_cand(float* __restrict__ ws)
{
    __shared__ float sred[256];
    __shared__ int   sidx[256];
    __shared__ int   s_sel;
    const int tid = threadIdx.x;
    const float gmax = ws[O_GMAX];
    const int stride = NB_SUM * 256;              // 131072
    const int i0 = blockIdx.x * 256 + tid;
    const int i1 = i0 + stride;
    const float v0 = ws[O_LOGITS + i0];
    const float v1 = ws[O_LOGITS + i1];

    // partial softmax denominator
    sred[tid] = expf(v0 - gmax) + expf(v1 - gmax);
    __syncthreads();
    for (int s = 128; s > 0; s >>= 1) {
        if (tid < s) sred[tid] += sred[tid + s];
        __syncthreads();
    }
    if (tid == 0) ws[O_BSUM + blockIdx.x] = sred[0];
    __syncthreads();

    // per-block top-NCAND via iterative argmax (values live in registers)
    int* wsi = (int*)ws;
    bool used0 = false, used1 = false;
    for (int p = 0; p < NCAND; ++p) {
        float bv = -FLT_MAX; int bi = 0x7fffffff;
        if (!used0) { bv = v0; bi = i0; }
        if (!used1 && (v1 > bv || (v1 == bv && i1 < bi))) { bv = v1; bi = i1; }
        sred[tid] = bv; sidx[tid] = bi;
        __syncthreads();
        for (int s = 128; s > 0; s >>= 1) {
            if (tid < s) {
                const float v2 = sred[tid + s]; const int i2 = sidx[tid + s];
                if (v2 > sred[tid] || (v2 == sred[tid] && i2 < sidx[tid])) {
                    sred[tid] = v2; sidx[tid] = i2;
                }
            }
            __syncthreads();
        }
        if (tid == 0) {
            ws [O_CANDV + blockIdx.x * NCAND + p] = sred[0];
            wsi[O_CANDI + blockIdx.x * NCAND + p] = sidx[0];
            s_sel = sidx[0];
        }
        __syncthreads();
        if (s_sel == i0) used0 = true;
        if (s_sel == i1) used1 = true;
        __syncthreads();
    }
}

// ---------------------------------------------------------------------------
// Kernel 5: finalize. denom; merge 4096 candidates -> top-k (lowest-index
// tie-break, matching jax.lax.top_k); scores; combined entity vector;
// Wv GEMV; Wo GEMV -> out. 1 block x 512 threads, touches ~570 KB total.
// ---------------------------------------------------------------------------
__global__ void evg_final(const float* __restrict__ E,
                          const float* __restrict__ Wv, const float* __restrict__ bv,
                          const float* __restrict__ Wo, const float* __restrict__ bo,
                          const int*   __restrict__ topk,
                          const float* __restrict__ ws, float* __restrict__ out)
{
    __shared__ float sred[512];
    __shared__ int   sidx[512];
    __shared__ int   ssel[NCAND];
    __shared__ float sscore[NCAND];
    __shared__ float secomb[D_IN];
    __shared__ float swsum[D_HID];
    __shared__ float s_denom, s_gmax;
    const int tid = threadIdx.x;                  // 0..511
    const int* wsi = (const int*)ws;
    int k = topk[0];
    if (k < 0) k = 0;
    if (k > NCAND) k = NCAND;

    // softmax denominator
    sred[tid] = ws[O_BSUM + tid];
    __syncthreads();
    for (int s = 256; s > 0; s >>= 1) {
        if (tid < s) sred[tid] += sred[tid + s];
        __syncthreads();
    }
    if (tid == 0) { s_denom = sred[0]; s_gmax = ws[O_GMAX]; }
    __syncthreads();

    // merge NB_SUM*NCAND = 4096 candidates -> top-k
    const int NC = NB_SUM * NCAND;
    for (int p = 0; p < k; ++p) {
        float bval = -FLT_MAX; int bidx = 0x7fffffff;
        for (int j = tid; j < NC; j += 512) {
            const int  ci = wsi[O_CANDI + j];
            bool skip = false;
            for (int pp = 0; pp < p; ++pp) if (ci == ssel[pp]) skip = true;
            if (skip) continue;
            const float cv = ws[O_CANDV + j];
            if (cv > bval || (cv == bval && ci < bidx)) { bval = cv; bidx = ci; }
        }
        sred[tid] = bval; sidx[tid] = bidx;
        __syncthreads();
        for (int s = 256; s > 0; s >>= 1) {
            if (tid < s) {
                const float v2 = sred[tid + s]; const int i2 = sidx[tid + s];
                if (v2 > sred[tid] || (v2 == sred[tid] && i2 < sidx[tid])) {
                    sred[tid] = v2; sidx[tid] = i2;
                }
            }
            __syncthreads();
        }
        if (tid == 0) ssel[p] = sidx[0];
        __syncthreads();
    }

    if (tid == 0) {
        for (int p = 0; p < k; ++p)
            sscore[p] = expf(ws[O_LOGITS + ssel[p]] - s_gmax) / s_denom;
    }
    __syncthreads();

    // combined entity vector: ecomb = sum_p score_p * E[idx_p, :]
    {
        float acc = 0.f;
        for (int p = 0; p < k; ++p)
            acc += sscore[p] * E[(size_t)ssel[p] * D_IN + tid];
        secomb[tid] = acc;
    }
    __syncthreads();

    // weighted_sum = Wv @ ecomb + (sum_p score_p) * bv
    if (tid < D_HID) {
        float ssum = 0.f;
        for (int p = 0; p < k; ++p) ssum += sscore[p];
        float acc = bv[tid] * ssum;
        const float* wr = Wv + tid * D_IN;
        for (int d = 0; d < D_IN; ++d) acc += wr[d] * secomb[d];
        swsum[tid] = acc;
    }
    __syncthreads();

    // out = Wo @ weighted_sum + bo
    {
        float acc = bo[tid];
        const float* wr = Wo + tid * D_HID;
        for (int h = 0; h < D_HID; ++h) acc += wr[h] * swsum[h];
        out[tid] = acc;
    }
}

// ---------------------------------------------------------------------------
extern "C" void kernel_launch(void* const* d_in, const int* in_sizes, int n_in,
                              void* d_out, int out_size, void* d_ws, size_t ws_size,
                              hipStream_t stream)
{
    const float* cemb = (const float*)d_in[0];   // class_embedding (512)
    const float* E    = (const float*)d_in[1];   // entity_embeddings (262144 x 512)
    const float* Wq   = (const float*)d_in[2];   // (128 x 512)
    const float* bq   = (const float*)d_in[3];   // (128)
    const float* Wk   = (const float*)d_in[4];   // (128 x 512)
    const float* bk   = (const float*)d_in[5];   // (128)
    const float* Wv   = (const float*)d_in[6];   // (128 x 512)
    const float* bv   = (const float*)d_in[7];   // (128)
    const float* Wo   = (const float*)d_in[8];   // (512 x 128)
    const float* bo   = (const float*)d_in[9];   // (512)
    const int*   topk = (const int*)d_in[10];    // scalar
    float* ws  = (float*)d_ws;
    float* out = (float*)d_out;

    evg_prep       <<<1,         512, 0, stream>>>(cemb, Wq, bq, Wk, bk, ws);
    evg_logits     <<<NB_LOGITS, 256, 0, stream>>>(E, ws);
    evg_gmax       <<<1,         512, 0, stream>>>(ws);
    evg_expsum_cand<<<NB_SUM,    256, 0, stream>>>(ws);
    evg_final      <<<1,         512, 0, stream>>>(E, Wv, bv, Wo, bo, topk, ws, out);
}